// DNBP_82446192214799
// MI455X (gfx1250) — compile-verified
//
#include <hip/hip_runtime.h>

#define B_ 8
#define N_ 20
#define K_ 2
#define P_ 320
#define D_ 3
#define F_ 64
#define H_ 64
#define KP_ (K_*P_)          // 640
#define EPSF 1e-8f

typedef float v2f __attribute__((ext_vector_type(2)));
typedef float v8f __attribute__((ext_vector_type(8)));

// One block per (b,n,k): 320 threads = 10 waves (wave32).
// Each wave owns 2 M-tiles of 16 rows; per tile:
//   40x V_WMMA_F32_16X16X4_F32 for cross (K-dim = D=3, padded to 4)
//    4x V_WMMA_F32_16X16X4_F32 for hx  (K-dim = D=3)
// B-fragments are pre-swizzled in LDS as {d0,d1,d2,0} quads so each lane's
// v2f operand is one unconditional ds_load_b64 (no exec-masked ternaries
// in the inner loop).
__global__ __launch_bounds__(320)
void dnbp_main(const float* __restrict__ X,   const float* __restrict__ W,
               const float* __restrict__ feats,const float* __restrict__ mu,
               const float* __restrict__ W1,  const float* __restrict__ Wx,
               const float* __restrict__ b1,  const float* __restrict__ W2,
               const float* __restrict__ bias2,const int* __restrict__ nbr,
               float* __restrict__ out, float* __restrict__ part)
{
    __shared__ __align__(16) float sB  [KP_*4];  // {xn0,xn1,xn2,0} per neighbor q
    __shared__ __align__(16) float sQW [KP_*2];  // {C2*|xn|^2, w_norm} per q
    __shared__ __align__(16) float sX  [P_*3];   // own points
    __shared__ __align__(16) float sA2 [P_];     // C2*|x_p - mu|^2
    __shared__ __align__(16) float sWxB[H_*4];   // {Wx0,Wx1,Wx2,0} per h
    __shared__ __align__(16) float sHW [H_*2];   // {hf+b1, W2} per h
    __shared__ __align__(16) float sRed[P_];     // deterministic tree reductions
    __shared__ float sSum;

    const int blk = blockIdx.x;          // ((b*N+n)*K)+k
    const int k   = blk % K_;
    const int bn  = blk / K_;
    const int n   = bn % N_;
    const int b   = bn / N_;
    const int t   = threadIdx.x;         // 0..319

    const int m = nbr[n*K_ + k];         // neighbor node id

    const float C2  = -2.885390081777927f; // -2*log2(e): kern = 2^(C2*d2)
    const float C4  = -2.0f * C2;          // +4*log2(e): coeff of cross
    const float L2E = 1.4426950408889634f;

    // ---------------- stage 1: fill LDS ----------------
    const float mu0 = mu[(n*K_+k)*D_ + 0];
    const float mu1 = mu[(n*K_+k)*D_ + 1];
    const float mu2 = mu[(n*K_+k)*D_ + 2];

    {   // own rows (one per thread)
        const float* xr = X + (size_t)((bn*K_ + k)*P_ + t)*D_;
        float x0 = xr[0], x1 = xr[1], x2 = xr[2];
        sX[t*3+0] = x0; sX[t*3+1] = x1; sX[t*3+2] = x2;
        float a0 = x0-mu0, a1 = x1-mu1, a2 = x2-mu2;
        sA2[t] = C2 * (a0*a0 + a1*a1 + a2*a2);
    }
    {   // neighbor rows: 2 per thread (KP=640), B-frags pre-swizzled
        const float* xb = X + (size_t)(b*N_+m)*KP_*D_;
        const float* wb = W + (size_t)(b*N_+m)*KP_;
        float wsl = 0.f;
        #pragma unroll
        for (int j = 0; j < 2; ++j) {
            int q = t + j*P_;
            float y0 = xb[q*3+0], y1 = xb[q*3+1], y2 = xb[q*3+2];
            sB[q*4+0] = y0; sB[q*4+1] = y1; sB[q*4+2] = y2; sB[q*4+3] = 0.f;
            sQW[q*2+0] = C2 * (y0*y0 + y1*y1 + y2*y2);
            float wv   = wb[q];
            sQW[q*2+1] = wv;           // normalized after wsum reduction
            wsl       += wv;
        }
        sRed[t] = wsl;
    }
    if (t < H_) {   // gating constants: hf = feats.W1 + b1 ; W2 ; Wx quads
        const float* fr = feats + (size_t)(b*N_+n)*F_;
        const float* w1 = W1    + (size_t)n*F_*H_;
        float acc = 0.f;
        #pragma unroll 8
        for (int f = 0; f < F_; ++f) acc += fr[f] * w1[f*H_ + t];
        sHW[t*2+0] = acc + b1[n*H_ + t];
        sHW[t*2+1] = W2[n*H_ + t];
        const float* wx = Wx + (size_t)n*D_*H_;
        sWxB[t*4+0] = wx[0*H_ + t];
        sWxB[t*4+1] = wx[1*H_ + t];
        sWxB[t*4+2] = wx[2*H_ + t];
        sWxB[t*4+3] = 0.f;
    }
    __syncthreads();

    // deterministic tree reduction of neighbor weight sum (320 partials)
    if (t < 160) sRed[t] += sRed[t+160]; __syncthreads();
    if (t <  80) sRed[t] += sRed[t+ 80]; __syncthreads();
    if (t <  40) sRed[t] += sRed[t+ 40]; __syncthreads();
    if (t <  20) sRed[t] += sRed[t+ 20]; __syncthreads();
    if (t <  10) sRed[t] += sRed[t+ 10]; __syncthreads();
    if (t == 0) { float s = 0.f; for (int i = 0; i < 10; ++i) s += sRed[i]; sSum = s; }
    __syncthreads();
    {   // normalize neighbor weights
        float inv = 1.0f / (sSum + EPSF);
        sQW[t*2+1]        *= inv;
        sQW[(t+P_)*2+1]   *= inv;
    }
    __syncthreads();

    // ---------------- stage 2: WMMA compute ----------------
    const int wave = t >> 5;
    const int lane = t & 31;
    const int col  = lane & 15;     // C-matrix column held by this lane
    const int half = lane >> 4;     // rows 0-7 vs 8-15 of tile

    const float  bz = bias2[n];
    const size_t ob = (size_t)blk * P_;

    // per-lane base pointers for unconditional b64 fragment loads
    const float* bBase  = &sB  [col*4 + half*2];  // + qt*64 per q-tile
    const float* qwBase = &sQW [col*2];           // + qt*32 per q-tile
    const float* wxBase = &sWxB[col*4 + half*2];  // + nt*64 per h-tile
    const float* hwBase = &sHW [col*2];           // + nt*32 per h-tile

    for (int it = 0; it < 2; ++it) {
        const int mt = wave + it*10;         // M-tile index (0..19)
        const int p0 = mt * 16;
        const int rowA = p0 + col;           // A-frag: lane -> row M = lane%16

        // A fragments (16x4 fp32): VGPR0 = K{0|2}, VGPR1 = K{1|3} by lane-half
        float x0 = sX[rowA*3+0], x1 = sX[rowA*3+1], x2 = sX[rowA*3+2];
        v2f av, xv;
        if (half == 0) { av.x = x0-mu0; av.y = x1-mu1; xv.x = x0; xv.y = x1; }
        else           { av.x = x2-mu2; av.y = 0.f;    xv.x = x2; xv.y = 0.f; }

        float ea[8];                         // C2*|a_p|^2 for this lane's 8 rows
        #pragma unroll
        for (int r = 0; r < 8; ++r) ea[r] = sA2[p0 + half*8 + r];

        float msg[8], ug[8];
        #pragma unroll
        for (int r = 0; r < 8; ++r) { msg[r] = 0.f; ug[r] = 0.f; }

        // ---- Gaussian kernel row-block x all 640 neighbor cols ----
        #pragma unroll 2
        for (int qt = 0; qt < 40; ++qt) {
            v2f bv = *(const v2f*)(bBase + qt*64);     // one ds_load_b64
            v8f c = {};
            c = __builtin_amdgcn_wmma_f32_16x16x4_f32(
                    false, av, false, bv, (short)0, c, false, false);
            v2f qw = *(const v2f*)(qwBase + qt*32);    // {exn, wq}
            #pragma unroll
            for (int r = 0; r < 8; ++r) {
                // exp(-2*d2) = 2^( C2*|a|^2 + C2*|xn|^2 + C4*cross )
                float arg = __builtin_fmaf(C4, c[r], ea[r] + qw.x);
                msg[r] = __builtin_fmaf(__builtin_amdgcn_exp2f(arg), qw.y, msg[r]);
            }
        }

        // ---- gating MLP: hx = X.Wx (4 H-tiles), relu, dot with W2 ----
        #pragma unroll
        for (int nt = 0; nt < 4; ++nt) {
            v2f wv = *(const v2f*)(wxBase + nt*64);    // one ds_load_b64
            v8f hx = {};
            hx = __builtin_amdgcn_wmma_f32_16x16x4_f32(
                    false, xv, false, wv, (short)0, hx, false, false);
            v2f hw = *(const v2f*)(hwBase + nt*32);    // {hf+b1, W2}
            #pragma unroll
            for (int r = 0; r < 8; ++r) {
                float hv = hx[r] + hw.x;
                hv = hv > 0.f ? hv : 0.f;
                ug[r] = __builtin_fmaf(hv, hw.y, ug[r]);
            }
        }

        // ---- reduce the 16 columns held across the 16-lane half ----
        #pragma unroll
        for (int mask = 1; mask < 16; mask <<= 1) {
            #pragma unroll
            for (int r = 0; r < 8; ++r) {
                msg[r] += __shfl_xor(msg[r], mask, 32);
                ug[r]  += __shfl_xor(ug[r],  mask, 32);
            }
        }
        if (col == 0) {   // lanes 0 (rows 0-7) and 16 (rows 8-15)
            #pragma unroll
            for (int r = 0; r < 8; ++r) {
                int   p = p0 + half*8 + r;
                // sigmoid(t) = rcp(1 + 2^(-log2(e)*t)); exact tail saturation
                float tgt = ug[r] + bz;
                float u = __builtin_amdgcn_rcpf(
                              1.0f + __builtin_amdgcn_exp2f(-L2E * tgt));
                float wval = u * msg[r];
                out[ob + p] = wval;      // unnormalized; kernel 2 divides
                sRed[p]     = wval;
            }
        }
    }
    __syncthreads();

    // deterministic block partial sum for the final (k,p) normalization
    if (t < 160) sRed[t] += sRed[t+160]; __syncthreads();
    if (t <  80) sRed[t] += sRed[t+ 80]; __syncthreads();
    if (t <  40) sRed[t] += sRed[t+ 40]; __syncthreads();
    if (t <  20) sRed[t] += sRed[t+ 20]; __syncthreads();
    if (t <  10) sRed[t] += sRed[t+ 10]; __syncthreads();
    if (t == 0) { float s = 0.f; for (int i = 0; i < 10; ++i) s += sRed[i]; part[blk] = s; }
}

__global__ __launch_bounds__(256)
void dnbp_norm(float* __restrict__ out, const float* __restrict__ part)
{
    int i = blockIdx.x * blockDim.x + threadIdx.x;
    if (i >= B_*N_*K_*P_) return;
    int bn = i / (K_*P_);
    float s = part[bn*K_ + 0] + part[bn*K_ + 1];
    out[i] = out[i] / (s + EPSF);
}

extern "C" void kernel_launch(void* const* d_in, const int* in_sizes, int n_in,
                              void* d_out, int out_size, void* d_ws, size_t ws_size,
                              hipStream_t stream)
{
    const float* X     = (const float*)d_in[0];
    const float* W     = (const float*)d_in[1];
    const float* feats = (const float*)d_in[2];
    const float* mu    = (const float*)d_in[3];
    const float* W1    = (const float*)d_in[4];
    const float* Wx    = (const float*)d_in[5];
    const float* b1    = (const float*)d_in[6];
    const float* W2    = (const float*)d_in[7];
    const float* bias2 = (const float*)d_in[8];
    const int*   nbr   = (const int*)d_in[9];
    float* out  = (float*)d_out;
    float* part = (float*)d_ws;   // B*N*K floats = 1280 bytes of scratch

    dnbp_main<<<B_*N_*K_, 320, 0, stream>>>(X, W, feats, mu, W1, Wx, b1, W2,
                                            bias2, nbr, out, part);
    dnbp_norm<<<(B_*N_*K_*P_ + 255)/256, 256, 0, stream>>>(out, part);
}